// Net_75694503624707
// MI455X (gfx1250) — compile-verified
//
#include <hip/hip_runtime.h>
#include <math.h>

// ---------------------------------------------------------------------------
// GAT-2-layer + 3 MLP heads for MI455X (gfx1250, wave32).
// Dense GEMMs use V_WMMA_F32_16X16X4_F32 (full f32 precision, matches ref).
// Sparse segment-softmax/aggregation uses L2-resident float atomics with
// compile-time (H,C) so the gathers vectorize to b128 loads.
// ---------------------------------------------------------------------------

typedef float v2f __attribute__((ext_vector_type(2)));
typedef float v8f __attribute__((ext_vector_type(8)));

#define NEG_SLOPE 0.2f

__device__ __forceinline__ void atomicMaxF32(float* addr, float v) {
  // Monotonic int encoding of IEEE floats: max(float) == max(int) for >=0,
  // == min(uint) for <0.  Init value is -INF (0xFF800000).
  if (v >= 0.0f) atomicMax((int*)addr, __float_as_int(v));
  else           atomicMin((unsigned int*)addr, __float_as_uint(v));
}

__device__ __forceinline__ float leaky(float v) {
  return v > 0.f ? v : NEG_SLOPE * v;
}

// ---- WMMA GEMM: Hout[N,64] = X[N,K] @ W[K,64], f32, no bias ----------------
// One wave computes one 16x16 output tile; 4 waves/block cover 16 rows x 64
// cols.  N must be a multiple of 16 (50000 = 3125*16), K a multiple of 4.
// A fragment (16x4 f32): lanes 0-15 hold K=0,1; lanes 16-31 hold K=2,3 (M=lane&15).
// B fragment (4x16 f32): lanes 0-15 hold K=0,1; lanes 16-31 hold K=2,3 (Ncol=lane&15).
// C/D (16x16 f32): VGPR i -> row i (lanes 0-15) / row i+8 (lanes 16-31).
__global__ void gemm_wmma_f32(const float* __restrict__ X,
                              const float* __restrict__ W,
                              float* __restrict__ Hout, int K) {
  const int lane    = threadIdx.x & 31;
  const int wave    = threadIdx.x >> 5;     // 0..3 -> column tile
  const int half    = lane >> 4;            // 0 | 1
  const int idx16   = lane & 15;
  const int rowbase = blockIdx.x * 16;
  const int colbase = wave * 16;
  const int arow    = rowbase + idx16;      // M for the A fragment
  const int bcol    = colbase + idx16;      // N for the B fragment

  v8f acc = {0.f, 0.f, 0.f, 0.f, 0.f, 0.f, 0.f, 0.f};
  for (int k = 0; k < K; k += 4) {
    const int ks = k + 2 * half;
    v2f a, b;
    a.x = X[(size_t)arow * K + ks + 0];
    a.y = X[(size_t)arow * K + ks + 1];
    b.x = W[(size_t)(ks + 0) * 64 + bcol];
    b.y = W[(size_t)(ks + 1) * 64 + bcol];
    acc = __builtin_amdgcn_wmma_f32_16x16x4_f32(
        /*neg_a=*/false, a, /*neg_b=*/false, b,
        /*c_mod=*/(short)0, acc, /*reuse_a=*/false, /*reuse_b=*/false);
  }
#pragma unroll
  for (int i = 0; i < 8; ++i)
    Hout[(size_t)(rowbase + i + 8 * half) * 64 + bcol] = acc[i];
}

// ---- attention coefficients: al[n,h]=sum_c Hm[n,h,c]*asrc[h,c]; ar likewise
template <int Hh, int C>
__global__ void attn_coeff(const float* __restrict__ Hm,
                           const float* __restrict__ asrc,
                           const float* __restrict__ adst,
                           float* __restrict__ al, float* __restrict__ ar,
                           int N) {
  int t = blockIdx.x * blockDim.x + threadIdx.x;
  if (t >= N * Hh) return;
  int n = t / Hh, h = t % Hh;
  const float* row = Hm + (size_t)n * (Hh * C) + (size_t)h * C;
  float sl = 0.f, sr = 0.f;
#pragma unroll
  for (int c = 0; c < C; ++c) {
    float v = row[c];
    sl += v * asrc[h * C + c];
    sr += v * adst[h * C + c];
  }
  al[t] = sl;
  ar[t] = sr;
}

// ---- fill ------------------------------------------------------------------
__global__ void fill_f32(float* __restrict__ p, float v, long n) {
  long t = (long)blockIdx.x * blockDim.x + threadIdx.x;
  if (t < n) p[t] = v;
}

__device__ __forceinline__ void edge_sd(const int* __restrict__ ei, int e,
                                        int E, int& s, int& d) {
  if (e < E) { s = ei[e]; d = ei[E + e]; } else { s = d = e - E; }
}

// ---- edge pass A: segment max of leaky_relu(al[s]+ar[d]) over dst ----------
template <int Hh>
__global__ void edge_max(const int* __restrict__ ei, int E, int N,
                         const float* __restrict__ al,
                         const float* __restrict__ ar,
                         float* __restrict__ emax) {
  int e = blockIdx.x * blockDim.x + threadIdx.x;
  if (e >= E + N) return;
  int s, d;
  edge_sd(ei, e, E, s, d);
#pragma unroll
  for (int h = 0; h < Hh; ++h)
    atomicMaxF32(&emax[d * Hh + h], leaky(al[s * Hh + h] + ar[d * Hh + h]));
}

// ---- edge pass B: z[d,h] += exp(e - emax[d,h]) -----------------------------
template <int Hh>
__global__ void edge_sum(const int* __restrict__ ei, int E, int N,
                         const float* __restrict__ al,
                         const float* __restrict__ ar,
                         const float* __restrict__ emax,
                         float* __restrict__ z) {
  int e = blockIdx.x * blockDim.x + threadIdx.x;
  if (e >= E + N) return;
  int s, d;
  edge_sd(ei, e, E, s, d);
#pragma unroll
  for (int h = 0; h < Hh; ++h) {
    float v = leaky(al[s * Hh + h] + ar[d * Hh + h]);
    atomicAdd(&z[d * Hh + h], expf(v - emax[d * Hh + h]));
  }
}

// ---- edge pass C: agg[d,h,:] += Hm[s,h,:] * alpha(e,h) ---------------------
// Compile-time (Hh,C) => the 64-float row gather vectorizes to 16x b128 loads.
template <int Hh, int C>
__global__ void edge_aggr(const int* __restrict__ ei, int E, int N,
                          const float* __restrict__ Hm,
                          const float* __restrict__ al,
                          const float* __restrict__ ar,
                          const float* __restrict__ emax,
                          const float* __restrict__ z,
                          float* __restrict__ agg) {
  int e = blockIdx.x * blockDim.x + threadIdx.x;
  if (e >= E + N) return;
  int s, d;
  edge_sd(ei, e, E, s, d);
  constexpr int HC = Hh * C;
#pragma unroll
  for (int h = 0; h < Hh; ++h) {
    float v = leaky(al[s * Hh + h] + ar[d * Hh + h]);
    float p = expf(v - emax[d * Hh + h]);
    float alpha = p / (z[d * Hh + h] + 1e-16f);
    const float4* srow4 =
        reinterpret_cast<const float4*>(Hm + (size_t)s * HC + h * C);
    float* drow = agg + (size_t)d * HC + h * C;
#pragma unroll
    for (int c4 = 0; c4 < C / 4; ++c4) {
      float4 v4 = srow4[c4];
      atomicAdd(&drow[c4 * 4 + 0], v4.x * alpha);
      atomicAdd(&drow[c4 * 4 + 1], v4.y * alpha);
      atomicAdd(&drow[c4 * 4 + 2], v4.z * alpha);
      atomicAdd(&drow[c4 * 4 + 3], v4.w * alpha);
    }
  }
}

// ---- bias + ELU, in place over [N,64] --------------------------------------
__global__ void bias_elu(float* __restrict__ a, const float* __restrict__ bias,
                         long n) {
  long t = (long)blockIdx.x * blockDim.x + threadIdx.x;
  if (t >= n) return;
  float v = a[t] + bias[t & 63];
  a[t] = v > 0.f ? v : expm1f(v);
}

// ---- 3 regression heads: out[n,i] = relu(hf@wa_i + ba_i) @ wb_i + bb_i -----
__global__ void mlp_heads(const float* __restrict__ hf,
                          const float* __restrict__ wa0, const float* __restrict__ ba0,
                          const float* __restrict__ wb0, const float* __restrict__ bb0,
                          const float* __restrict__ wa1, const float* __restrict__ ba1,
                          const float* __restrict__ wb1, const float* __restrict__ bb1,
                          const float* __restrict__ wa2, const float* __restrict__ ba2,
                          const float* __restrict__ wb2, const float* __restrict__ bb2,
                          float* __restrict__ out, int N) {
  int n = blockIdx.x * blockDim.x + threadIdx.x;
  if (n >= N) return;
  float h[64];
  const float4* hrow4 = reinterpret_cast<const float4*>(hf + (size_t)n * 64);
#pragma unroll
  for (int i = 0; i < 16; ++i) {
    float4 v4 = hrow4[i];
    h[i * 4 + 0] = v4.x; h[i * 4 + 1] = v4.y;
    h[i * 4 + 2] = v4.z; h[i * 4 + 3] = v4.w;
  }
  const float* WA[3] = {wa0, wa1, wa2};
  const float* BA[3] = {ba0, ba1, ba2};
  const float* WB[3] = {wb0, wb1, wb2};
  const float* BB[3] = {bb0, bb1, bb2};
  for (int t = 0; t < 3; ++t) {
    float r = BB[t][0];
    for (int j = 0; j < 32; ++j) {
      float s = BA[t][j];
#pragma unroll
      for (int k = 0; k < 64; ++k) s += h[k] * WA[t][(size_t)k * 32 + j];
      r += fmaxf(s, 0.f) * WB[t][j];
    }
    out[(size_t)n * 3 + t] = r;
  }
}

// ---------------------------------------------------------------------------
extern "C" void kernel_launch(void* const* d_in, const int* in_sizes, int n_in,
                              void* d_out, int out_size, void* d_ws, size_t ws_size,
                              hipStream_t stream) {
  const float* x   = (const float*)d_in[0];
  const int*   ei  = (const int*)d_in[1];
  const float* W1  = (const float*)d_in[2];
  const float* as1 = (const float*)d_in[3];
  const float* ad1 = (const float*)d_in[4];
  const float* b1  = (const float*)d_in[5];
  const float* W2  = (const float*)d_in[6];
  const float* as2 = (const float*)d_in[7];
  const float* ad2 = (const float*)d_in[8];
  const float* b2  = (const float*)d_in[9];
  const float* w15_1 = (const float*)d_in[10]; const float* b15_1 = (const float*)d_in[11];
  const float* w15_2 = (const float*)d_in[12]; const float* b15_2 = (const float*)d_in[13];
  const float* w30_1 = (const float*)d_in[14]; const float* b30_1 = (const float*)d_in[15];
  const float* w30_2 = (const float*)d_in[16]; const float* b30_2 = (const float*)d_in[17];
  const float* w45_1 = (const float*)d_in[18]; const float* b45_1 = (const float*)d_in[19];
  const float* w45_2 = (const float*)d_in[20]; const float* b45_2 = (const float*)d_in[21];
  float* out = (float*)d_out;

  const int N  = in_sizes[0] / 128;   // 50000 (multiple of 16)
  const int E  = in_sizes[1] / 2;     // 800000
  const int Et = E + N;               // edges incl. self-loops

  // workspace carve-out (floats); all block offsets keep 16B alignment
  float* p = (float*)d_ws;
  float* h1    = p; p += (size_t)N * 64;
  float* al1   = p; p += (size_t)N * 8;
  float* ar1   = p; p += (size_t)N * 8;
  float* emax1 = p; p += (size_t)N * 8;
  float* z1    = p; p += (size_t)N * 8;
  float* agg1  = p; p += (size_t)N * 64;   // becomes x2 after bias+ELU
  float* h2    = p; p += (size_t)N * 64;
  float* al2   = p; p += (size_t)N;
  float* ar2   = p; p += (size_t)N;
  float* emax2 = p; p += (size_t)N;
  float* z2    = p; p += (size_t)N;
  float* agg2  = p; p += (size_t)N * 64;   // becomes hf after bias+ELU

  const int TB = 256;
  const int gEdge = (Et + TB - 1) / TB;
  auto gN = [&](long n) { return (int)((n + TB - 1) / TB); };

  // ---------------- layer 1 (H=8, C=8) ----------------
  gemm_wmma_f32<<<N / 16, 128, 0, stream>>>(x, W1, h1, 128);
  attn_coeff<8, 8><<<gN((long)N * 8), TB, 0, stream>>>(h1, as1, ad1, al1, ar1, N);
  fill_f32<<<gN((long)N * 8),  TB, 0, stream>>>(emax1, -INFINITY, (long)N * 8);
  fill_f32<<<gN((long)N * 8),  TB, 0, stream>>>(z1, 0.f, (long)N * 8);
  fill_f32<<<gN((long)N * 64), TB, 0, stream>>>(agg1, 0.f, (long)N * 64);
  edge_max<8>    <<<gEdge, TB, 0, stream>>>(ei, E, N, al1, ar1, emax1);
  edge_sum<8>    <<<gEdge, TB, 0, stream>>>(ei, E, N, al1, ar1, emax1, z1);
  edge_aggr<8, 8><<<gEdge, TB, 0, stream>>>(ei, E, N, h1, al1, ar1, emax1, z1, agg1);
  bias_elu<<<gN((long)N * 64), TB, 0, stream>>>(agg1, b1, (long)N * 64);

  // ---------------- layer 2 (H=1, C=64) ----------------
  gemm_wmma_f32<<<N / 16, 128, 0, stream>>>(agg1, W2, h2, 64);
  attn_coeff<1, 64><<<gN((long)N), TB, 0, stream>>>(h2, as2, ad2, al2, ar2, N);
  fill_f32<<<gN((long)N),      TB, 0, stream>>>(emax2, -INFINITY, (long)N);
  fill_f32<<<gN((long)N),      TB, 0, stream>>>(z2, 0.f, (long)N);
  fill_f32<<<gN((long)N * 64), TB, 0, stream>>>(agg2, 0.f, (long)N * 64);
  edge_max<1>     <<<gEdge, TB, 0, stream>>>(ei, E, N, al2, ar2, emax2);
  edge_sum<1>     <<<gEdge, TB, 0, stream>>>(ei, E, N, al2, ar2, emax2, z2);
  edge_aggr<1, 64><<<gEdge, TB, 0, stream>>>(ei, E, N, h2, al2, ar2, emax2, z2, agg2);
  bias_elu<<<gN((long)N * 64), TB, 0, stream>>>(agg2, b2, (long)N * 64);

  // ---------------- regression heads ----------------
  mlp_heads<<<gN((long)N), TB, 0, stream>>>(agg2,
      w15_1, b15_1, w15_2, b15_2,
      w30_1, b30_1, w30_2, b30_2,
      w45_1, b45_1, w45_2, b45_2,
      out, N);
}